// ProposalLayer_17300128268727
// MI455X (gfx1250) — compile-verified
//
#include <hip/hip_runtime.h>
#include <hip/hip_bf16.h>
#include <stdint.h>

#define HH 83
#define WW 25
#define AA 9
#define N_ANCH (HH*WW*AA)      /* 18675 */
#define BATCH 16
#define PRE 12000
#define POST 2000
#define NPAD1 32768
#define NPAD2 16384
#define IMG_X 1333.0f
#define IMG_Y 402.0f
#define MIN_SZ 16.0f
#define NMS_TH 0.7f
#define INVALID_IDX 0xFFFFFFFFu

typedef unsigned int v4u  __attribute__((ext_vector_type(4)));
typedef int          v8i_t __attribute__((ext_vector_type(8)));
typedef int          v4i_t __attribute__((ext_vector_type(4)));

// ---------------- Kernel 1: decode boxes + build sort keys ----------------
__global__ void decode_kernel(const float* __restrict__ anchors,
                              const float* __restrict__ cls,
                              const float* __restrict__ pred,
                              float* __restrict__ roi,
                              unsigned long long* __restrict__ keys)
{
    int i = blockIdx.x * blockDim.x + threadIdx.x;
    if (i >= BATCH * NPAD1) return;
    int b = i / NPAD1, a = i % NPAD1;
    unsigned long long key = ~0ull;
    if (a < N_ANCH) {
        float ax1 = anchors[a*4+0], ay1 = anchors[a*4+1];
        float ax2 = anchors[a*4+2], ay2 = anchors[a*4+3];
        float ah = ay2 - ay1, aw = ax2 - ax1;
        float acy = ay1 + 0.5f*ah, acx = ax1 + 0.5f*aw;
        const float* pp = pred + ((size_t)b*N_ANCH + a)*4;
        float dx = pp[0], dy = pp[1], dw = pp[2], dh = pp[3];
        float cy = dy*ah + acy, cx = dx*aw + acx;
        float h = __expf(dh)*ah, w = __expf(dw)*aw;
        float x1 = fminf(fmaxf(cx - 0.5f*w, 0.f), IMG_X);
        float y1 = fminf(fmaxf(cy - 0.5f*h, 0.f), IMG_Y);
        float x2 = fminf(fmaxf(cx + 0.5f*w, 0.f), IMG_X);
        float y2 = fminf(fmaxf(cy + 0.5f*h, 0.f), IMG_Y);
        float* rp = roi + ((size_t)b*N_ANCH + a)*4;
        rp[0] = x1; rp[1] = y1; rp[2] = x2; rp[3] = y2;
        bool valid = ((y2 - y1) >= MIN_SZ) && ((x2 - x1) >= MIN_SZ);
        unsigned u;
        if (valid) u = __float_as_uint(cls[(size_t)b*(N_ANCH*2) + 2*a + 1]);
        else       u = 0xFF800000u;                       // -inf
        unsigned m = (u & 0x80000000u) ? ~u : (u | 0x80000000u);
        key = ((unsigned long long)(~m) << 32) | (unsigned)a; // asc = desc-score, idx tiebreak
    }
    keys[(size_t)b*NPAD1 + a] = key;
}

// ---------- Kernel 2: per-batch bitonic sort (256KB LDS) + gather + perm ----------
__global__ void __launch_bounds__(1024) sort_kernel(
    const float* __restrict__ roi,
    unsigned long long* __restrict__ keys,
    float* __restrict__ sBoxes,
    unsigned* __restrict__ aIdx,
    unsigned short* __restrict__ perm)
{
    extern __shared__ unsigned long long sk[];   // 32768 u64 = 256 KB (CDNA5 320KB LDS)
    const int NT = 1024;
    int b = blockIdx.x, tid = threadIdx.x;
    unsigned long long* gk = keys + (size_t)b*NPAD1;
    for (int i = tid; i < NPAD1; i += NT) sk[i] = gk[i];
    __syncthreads();
    // bitonic sort ascending, 32768 elements
    for (int k = 2; k <= NPAD1; k <<= 1)
        for (int j = k >> 1; j > 0; j >>= 1) {
            for (int t = tid; t < NPAD1/2; t += NT) {
                int i  = ((t & ~(j-1)) << 1) | (t & (j-1));
                int ix = i | j;
                unsigned long long a = sk[i], c = sk[ix];
                bool up = ((i & k) == 0);
                if ((a > c) == up) { sk[i] = c; sk[ix] = a; }
            }
            __syncthreads();
        }
    // cache my slice of the top-PRE keys
    unsigned long long myk[12];
    int cnt = 0;
    for (int p = tid; p < PRE; p += NT) myk[cnt++] = sk[p];
    __syncthreads();
    // emit sorted boxes / anchor ids; build perm keys in-place
    cnt = 0;
    for (int p = tid; p < PRE; p += NT) {
        unsigned long long kk = myk[cnt++];
        unsigned hi = (unsigned)(kk >> 32);
        unsigned a  = (unsigned)kk;
        bool valid = (hi != 0xFF800000u);
        const float4 box = *(const float4*)(roi + ((size_t)b*N_ANCH + a)*4);
        *(float4*)(sBoxes + ((size_t)b*PRE + p)*4) = box;
        aIdx[(size_t)b*PRE + p] = valid ? a : INVALID_IDX;
        int seq = valid ? (int)a : -1;
        unsigned hi2 = (unsigned)(0x7FFFFFFF - seq);     // asc = desc seq_key, stable in p
        sk[p] = ((unsigned long long)hi2 << 32) | (unsigned)p;
    }
    for (int p = PRE + tid; p < NPAD2; p += NT) sk[p] = ~0ull;
    __syncthreads();
    for (int k = 2; k <= NPAD2; k <<= 1)
        for (int j = k >> 1; j > 0; j >>= 1) {
            for (int t = tid; t < NPAD2/2; t += NT) {
                int i  = ((t & ~(j-1)) << 1) | (t & (j-1));
                int ix = i | j;
                unsigned long long a = sk[i], c = sk[ix];
                bool up = ((i & k) == 0);
                if ((a > c) == up) { sk[i] = c; sk[ix] = a; }
            }
            __syncthreads();
        }
    for (int r = tid; r < PRE; r += NT)
        perm[(size_t)b*PRE + r] = (unsigned short)(unsigned)sk[r];
}

// ---- Tensor Data Mover: 1-D tile DMA global->LDS (D# per CDNA5 ISA) ----
// dsizeCode: 0=1B, 1=2B, 2=4B, 3=8B.  nelem must fit tile_dim0 (16-bit).
__device__ __forceinline__ void tdm_load_1d(unsigned ldsOff, const void* gptr,
                                            unsigned nelem, unsigned dsizeCode)
{
    unsigned long long ga = (unsigned long long)(uintptr_t)gptr;
    v4u g0;
    g0.x = 1u;                                              // count=1, user mode
    g0.y = ldsOff;                                          // lds_addr (bytes)
    g0.z = (unsigned)(ga & 0xFFFFFFFFull);                  // global_addr[31:0]
    g0.w = (unsigned)((ga >> 32) & 0x1FFFFFFull) | (2u << 30); // addr[56:32] | type=2
    v8i_t g1;
    g1[0] = (int)(dsizeCode << 16);                         // wg_mask=0 | data_size
    g1[1] = (int)((nelem & 0xFFFFu) << 16);                 // tensor_dim0[15:0]
    g1[2] = (int)((nelem >> 16) | (1u << 16));              // tensor_dim0[31:16] | tensor_dim1=1
    g1[3] = (int)((nelem & 0xFFFFu) << 16);                 // tile_dim0 = nelem
    g1[4] = 0;                                              // tile_dim1/2 unused
    g1[5] = (int)nelem;                                     // tensor_dim0_stride[31:0]
    g1[6] = 0;
    g1[7] = 0;
    v4i_t g2 = {0,0,0,0}, g3 = {0,0,0,0};
    v8i_t g4 = {0,0,0,0,0,0,0,0};
    __builtin_amdgcn_tensor_load_to_lds(g0, g1, g2, g3, g4, 0);
}

// ---------------- Kernel 3: NMS, whole working set LDS-resident ----------------
// LDS layout (~300 KB, fits CDNA5's 320KB WGP LDS):
//   [0,192000)        : boxes float4[PRE]   (TDM)
//   [192000,216000)   : perm  u16[PRE]      (TDM)
//   [216000,240000)   : rankOf u16[PRE]
//   [240000,252000)   : alive u8[PRE]       (indexed by rank)
//   [252000,252132)   : reduction scratch u32[33]
//   [252160,300160)   : anchor-id/validity u32[PRE] (TDM)
__global__ void __launch_bounds__(1024) nms_kernel(
    const float* __restrict__ sBoxes,
    const unsigned* __restrict__ aIdx,
    const unsigned short* __restrict__ perm,
    float* __restrict__ outRois,
    float* __restrict__ outMask)
{
    extern __shared__ char smem[];
    float*          bx    = (float*)smem;
    unsigned short* sp    = (unsigned short*)(smem + 192000);
    unsigned short* srank = (unsigned short*)(smem + 216000);
    unsigned char*  alive = (unsigned char*)(smem + 240000);
    unsigned*       red   = (unsigned*)(smem + 252000);
    unsigned*       sAid  = (unsigned*)(smem + 252160);
    const int NT = 1024;
    int b = blockIdx.x, tid = threadIdx.x;

    // One wave issues all three TDM descriptors (in-order per wave), single wait.
    if (tid < 32) {
        tdm_load_1d((unsigned)(uintptr_t)(void*)bx,
                    sBoxes + (size_t)b*PRE*4, PRE*4, 2u);   // 192000 B f32
        tdm_load_1d((unsigned)(uintptr_t)(void*)sp,
                    perm + (size_t)b*PRE,     PRE,   1u);   //  24000 B u16
        tdm_load_1d((unsigned)(uintptr_t)(void*)sAid,
                    aIdx + (size_t)b*PRE,     PRE,   2u);   //  48000 B u32
        __builtin_amdgcn_s_wait_tensorcnt(0);
    }
    __syncthreads();

    for (int r = tid; r < PRE; r += NT)
        srank[sp[r]] = (unsigned short)r;
    __syncthreads();
    for (int r = tid; r < PRE; r += NT)
        alive[r] = (sAid[sp[r]] != INVALID_IDX) ? 1 : 0;
    __syncthreads();

    float* roisB = outRois + (size_t)b*POST*4;
    float* maskB = outMask + (size_t)b*POST;
    int cursor = 0;
    int it = 0;
    for (; it < POST; ++it) {
        // first alive rank >= cursor (alive ranks only die -> monotone cursor)
        unsigned rsel = 0xFFFFFFFFu;
        int base = cursor;
        while (base < PRE) {
            int mine = base + tid;
            unsigned v = (mine < PRE && alive[mine]) ? (unsigned)mine : 0xFFFFFFFFu;
            __syncthreads();
            for (int off = 16; off > 0; off >>= 1) {
                unsigned o = (unsigned)__shfl_down(v, off, 32);
                v = (o < v) ? o : v;
            }
            if ((tid & 31) == 0) red[tid >> 5] = v;
            __syncthreads();
            if (tid < 32) {
                unsigned w = red[tid];
                for (int off = 16; off > 0; off >>= 1) {
                    unsigned o = (unsigned)__shfl_down(w, off, 32);
                    w = (o < w) ? o : w;
                }
                if (tid == 0) red[32] = w;
            }
            __syncthreads();
            rsel = red[32];
            if (rsel != 0xFFFFFFFFu) break;
            base += NT;
        }
        if (rsel == 0xFFFFFFFFu) break;          // nothing alive: rest is padding
        int r = (int)rsel;
        int j = (int)sp[r];
        float4 bj = *(float4*)(bx + j*4);
        float areaj = (bj.z - bj.x + 1.f) * (bj.w - bj.y + 1.f);
        for (int p = tid; p < PRE; p += NT) {
            float4 bp = *(float4*)(bx + p*4);
            float xx1 = fmaxf(bj.x, bp.x);
            float yy1 = fmaxf(bj.y, bp.y);
            float xx2 = fminf(bj.z, bp.z);
            float yy2 = fminf(bj.w, bp.w);
            float inter = fmaxf(0.f, xx2 - xx1 + 1.f) * fmaxf(0.f, yy2 - yy1 + 1.f);
            float areap = (bp.z - bp.x + 1.f) * (bp.w - bp.y + 1.f);
            float ovr = inter / (areaj + areap - inter);
            if (ovr > NMS_TH) alive[srank[p]] = 0;   // kills j itself too (IoU=1)
        }
        if (tid == 0) {
            *(float4*)(roisB + (size_t)it*4) = bj;
            maskB[it] = 1.0f;
        }
        cursor = r + 1;
        __syncthreads();
    }
    // zero-fill remaining slots
    for (int q = it*4 + tid; q < POST*4; q += NT) roisB[q] = 0.f;
    for (int q = it   + tid; q < POST;   q += NT) maskB[q] = 0.f;
}

// ------------------------------- launch -------------------------------
extern "C" void kernel_launch(void* const* d_in, const int* in_sizes, int n_in,
                              void* d_out, int out_size, void* d_ws, size_t ws_size,
                              hipStream_t stream)
{
    const float* anchors = (const float*)d_in[0];   // (N_ANCH,4)
    const float* cls     = (const float*)d_in[1];   // (B, N_ANCH*2)
    const float* pred    = (const float*)d_in[2];   // (B, N_ANCH, 4)

    char* ws = (char*)d_ws;
    unsigned long long* keys = (unsigned long long*)(ws);          // 4,194,304 B
    float*          roi      = (float*)(ws + 4194304);             // 4,780,800 B
    float*          sBoxes   = (float*)(ws + 8975104);             // 3,072,000 B
    unsigned*       aIdxArr  = (unsigned*)(ws + 12047104);         //   768,000 B
    unsigned short* permArr  = (unsigned short*)(ws + 12815104);   //   384,000 B

    float* outRois = (float*)d_out;                       // (B, POST, 4)
    float* outMask = outRois + (size_t)BATCH*POST*4;      // (B, POST)

    // opt in to >64KB dynamic LDS (CDNA5 WGP has 320KB)
    (void)hipFuncSetAttribute((const void*)sort_kernel,
                              hipFuncAttributeMaxDynamicSharedMemorySize, NPAD1*8);
    (void)hipFuncSetAttribute((const void*)nms_kernel,
                              hipFuncAttributeMaxDynamicSharedMemorySize, 300160);

    decode_kernel<<<(BATCH*NPAD1)/256, 256, 0, stream>>>(anchors, cls, pred, roi, keys);
    sort_kernel<<<BATCH, 1024, NPAD1*8, stream>>>(roi, keys, sBoxes, aIdxArr, permArr);
    nms_kernel<<<BATCH, 1024, 300160, stream>>>(sBoxes, aIdxArr, permArr, outRois, outMask);
}